// SAConv_63874753626416
// MI455X (gfx1250) — compile-verified
//
#include <hip/hip_runtime.h>
#include <hip/hip_bf16.h>

// ---------------------------------------------------------------------------
// SAConv on MI455X (gfx1250, wave32)
//  Stage 1: grouped 1x1 convs (q,k,v) via V_WMMA_F32_16X16X4_F32.
//           Tile = 16 pixels (M) x 16 channels (two heads, N) with K-chained
//           accumulation over ci (4 WMMAs per head-pair per matrix).
//  Stage 2: 7x7 windowed attention, LDS-tiled k/v, rel-pos folded into a
//           per-pixel q·rel bias (exact, handles zero-padding implicitly).
// ---------------------------------------------------------------------------

typedef __attribute__((ext_vector_type(2))) float v2f;
typedef __attribute__((ext_vector_type(8))) float v8f;

#define B_   8
#define HW_  64
#define CIN_ 64
#define COUT_ 64
#define HEADS_ 8
#define DH_  8
#define KS_  7
#define K2_  49
#define PAD_ 3
#define PIX_PER_IMG (HW_ * HW_)          // 4096
#define TOTPIX (B_ * PIX_PER_IMG)        // 32768

static __device__ __forceinline__ v8f wmma_f32x4(v2f a, v2f b, v8f c) {
    // D = A(16x4) x B(4x16) + C(16x16), all f32, wave32
    return __builtin_amdgcn_wmma_f32_16x16x4_f32(
        /*neg_a=*/false, a, /*neg_b=*/false, b,
        /*c_mod=*/(short)0, c, /*reuse_a=*/false, /*reuse_b=*/false);
}

// B fragment for the grouped conv: head h occupies N columns [colsel*8, colsel*8+8).
// Lane layout (assumed, mirrors the ISA A/C striping):
//   lanes 0-15  : x = B[kk+0][n], y = B[kk+1][n]
//   lanes 16-31 : x = B[kk+2][n], y = B[kk+3][n]
// w layout: (g, co=8, ci=8) row-major.
static __device__ __forceinline__ v2f load_bfrag(const float* __restrict__ w,
                                                 int h, int kk, int colsel, int lane) {
    const int n = lane & 15;
    const int krow = (lane >> 4) * 2;
    const bool act = colsel ? (n >= 8) : (n < 8);
    const int co = (colsel ? (n - 8) : n) & 7;      // clamped: load always in-bounds
    const float* p = w + h * 64 + co * 8 + kk + krow;
    v2f r;
    r.x = act ? p[0] : 0.0f;
    r.y = act ? p[1] : 0.0f;
    return r;
}

// ---------------------------------------------------------------------------
// Stage 1: q/k/v grouped 1x1 conv.  One wave handles 16 consecutive pixels
// (never crossing a row: 64 % 16 == 0).  Output channel-last: (B,H,W,64).
// Grid: 256 blocks x 256 threads = 2048 waves = 32768 pixels.
// ---------------------------------------------------------------------------
__global__ __launch_bounds__(256) void saconv_qkv_wmma(
    const float* __restrict__ x,     // (B,64,64,64)
    const float* __restrict__ wq,
    const float* __restrict__ wk,
    const float* __restrict__ wv,
    float* __restrict__ q_ws,        // (B,H,W,64)
    float* __restrict__ k_ws,
    float* __restrict__ v_ws)
{
    const int lane  = threadIdx.x & 31;
    const int wave  = blockIdx.x * 8 + (threadIdx.x >> 5);   // 0..2047
    const int P     = wave * 16;                             // pixel base
    const int b     = P >> 12;
    const int ij    = P & 4095;                              // i*64 + j0

    const int m     = lane & 15;         // A-matrix row (pixel within group)
    const int khalf = lane >> 4;         // 0 -> holds K0,K1 ; 1 -> holds K2,K3

    const float* xb = x + (size_t)b * CIN_ * PIX_PER_IMG + ij + m;

    for (int hp = 0; hp < 4; ++hp) {
        const int h0 = hp * 2;
        const int h1 = h0 + 1;

        // A fragments: x[pixel][channel], channel = h*8 + kkbase + k
        v2f a00, a04, a10, a14;
        {
            int c;
            c = h0 * 8 + 0 + khalf * 2;
            a00.x = xb[(size_t)c * PIX_PER_IMG]; a00.y = xb[(size_t)(c + 1) * PIX_PER_IMG];
            c = h0 * 8 + 4 + khalf * 2;
            a04.x = xb[(size_t)c * PIX_PER_IMG]; a04.y = xb[(size_t)(c + 1) * PIX_PER_IMG];
            c = h1 * 8 + 0 + khalf * 2;
            a10.x = xb[(size_t)c * PIX_PER_IMG]; a10.y = xb[(size_t)(c + 1) * PIX_PER_IMG];
            c = h1 * 8 + 4 + khalf * 2;
            a14.x = xb[(size_t)c * PIX_PER_IMG]; a14.y = xb[(size_t)(c + 1) * PIX_PER_IMG];
        }

        // three output matrices (q,k,v), 4 chained WMMAs each
        v8f accq = {}, acck = {}, accv = {};
        accq = wmma_f32x4(a00, load_bfrag(wq, h0, 0, 0, lane), accq);
        accq = wmma_f32x4(a04, load_bfrag(wq, h0, 4, 0, lane), accq);
        accq = wmma_f32x4(a10, load_bfrag(wq, h1, 0, 1, lane), accq);
        accq = wmma_f32x4(a14, load_bfrag(wq, h1, 4, 1, lane), accq);

        acck = wmma_f32x4(a00, load_bfrag(wk, h0, 0, 0, lane), acck);
        acck = wmma_f32x4(a04, load_bfrag(wk, h0, 4, 0, lane), acck);
        acck = wmma_f32x4(a10, load_bfrag(wk, h1, 0, 1, lane), acck);
        acck = wmma_f32x4(a14, load_bfrag(wk, h1, 4, 1, lane), acck);

        accv = wmma_f32x4(a00, load_bfrag(wv, h0, 0, 0, lane), accv);
        accv = wmma_f32x4(a04, load_bfrag(wv, h0, 4, 0, lane), accv);
        accv = wmma_f32x4(a10, load_bfrag(wv, h1, 0, 1, lane), accv);
        accv = wmma_f32x4(a14, load_bfrag(wv, h1, 4, 1, lane), accv);

        // D layout: lane n (<16): VGPR r = D[M=r][N=n]; lane 16+n: D[M=8+r][N=n]
        const size_t base = (size_t)(P + khalf * 8) * COUT_ + hp * 16 + (lane & 15);
        float* dq = q_ws + base;
        float* dk = k_ws + base;
        float* dv = v_ws + base;
#pragma unroll
        for (int r = 0; r < 8; ++r) {
            dq[(size_t)r * COUT_] = accq[r];
            dk[(size_t)r * COUT_] = acck[r];
            dv[(size_t)r * COUT_] = accv[r];
        }
    }
}

// ---------------------------------------------------------------------------
// Stage 2: windowed attention.  One block = 16x16 pixel tile, one (b, head).
// LDS holds 22x22x8 k and v (OOB window positions = zeros => k_raw=0, v=0;
// their logits reduce to scale*bias which is exact for zero padding).
// Grid: B*HEADS*16 = 1024 blocks x 256 threads.
// ---------------------------------------------------------------------------
#define EXT_ 22

__global__ __launch_bounds__(256) void saconv_attn(
    const float* __restrict__ q_ws,   // (B,H,W,64)
    const float* __restrict__ k_ws,
    const float* __restrict__ v_ws,
    const float* __restrict__ rel_h,  // (32,7)
    const float* __restrict__ rel_w,  // (32,7)
    float* __restrict__ out)          // (B,64,H,W)
{
    __shared__ __align__(16) float ks_[EXT_ * EXT_ * DH_];
    __shared__ __align__(16) float vs_[EXT_ * EXT_ * DH_];

    const int blk  = blockIdx.x;
    const int tile = blk & 15;
    const int h    = (blk >> 4) & 7;
    const int b    = blk >> 7;
    const int i0   = (tile >> 2) * 16;
    const int j0   = (tile & 3) * 16;

    const int tid = threadIdx.x;
    const int px  = tid & 15;
    const int py  = tid >> 4;

    const size_t bstride = (size_t)PIX_PER_IMG * COUT_;
    const float* kb = k_ws + (size_t)b * bstride;
    const float* vb = v_ws + (size_t)b * bstride;

    // cooperative tile fill (zeros outside the image)
    for (int idx = tid; idx < EXT_ * EXT_; idx += 256) {
        const int r  = idx / EXT_;
        const int c  = idx - r * EXT_;
        const int gi = i0 - PAD_ + r;
        const int gj = j0 - PAD_ + c;
        float4 k0 = make_float4(0.f, 0.f, 0.f, 0.f), k1 = k0, v0 = k0, v1 = k0;
        if (gi >= 0 && gi < HW_ && gj >= 0 && gj < HW_) {
            const size_t poff = ((size_t)gi * HW_ + gj) * COUT_ + h * DH_;
            k0 = *(const float4*)(kb + poff);
            k1 = *(const float4*)(kb + poff + 4);
            v0 = *(const float4*)(vb + poff);
            v1 = *(const float4*)(vb + poff + 4);
        }
        float* kd = ks_ + idx * DH_;
        float* vd = vs_ + idx * DH_;
        *(float4*)kd = k0; *(float4*)(kd + 4) = k1;
        *(float4*)vd = v0; *(float4*)(vd + 4) = v1;
    }

    // this thread's q vector
    float qr[DH_];
    {
        const float* qp = q_ws + (size_t)b * bstride +
                          ((size_t)(i0 + py) * HW_ + (j0 + px)) * COUT_ + h * DH_;
        const float4 q0 = *(const float4*)qp;
        const float4 q1 = *(const float4*)(qp + 4);
        qr[0] = q0.x; qr[1] = q0.y; qr[2] = q0.z; qr[3] = q0.w;
        qr[4] = q1.x; qr[5] = q1.y; qr[6] = q1.z; qr[7] = q1.w;
    }

    // rel-pos folded into a per-pixel bias: bias[d] = sum_c q[c]*rel[ch][d]
    // heads 0..3 use rel_h (along di), heads 4..7 use rel_w (along dj)
    float bias[KS_];
    {
        const float* rel = (h < 4) ? (rel_h + h * DH_ * KS_)
                                   : (rel_w + (h - 4) * DH_ * KS_);
#pragma unroll
        for (int d = 0; d < KS_; ++d) {
            float s = 0.f;
#pragma unroll
            for (int c = 0; c < DH_; ++c) s += qr[c] * rel[c * KS_ + d];
            bias[d] = s;
        }
    }

    __syncthreads();

    const float scale = 0.35355339059327373f;   // sqrt(heads/Cout)
    float lg[K2_];
    float mx = -3.0e38f;
    {
        int o = 0;
#pragma unroll
        for (int di = 0; di < KS_; ++di) {
#pragma unroll
            for (int dj = 0; dj < KS_; ++dj, ++o) {
                const float* kp = ks_ + (((py + di) * EXT_) + (px + dj)) * DH_;
                float dot = 0.f;
#pragma unroll
                for (int c = 0; c < DH_; ++c) dot += qr[c] * kp[c];
                const float l = scale * (dot + ((h < 4) ? bias[di] : bias[dj]));
                lg[o] = l;
                mx = fmaxf(mx, l);
            }
        }
    }

    float ssum = 0.f;
#pragma unroll
    for (int o = 0; o < K2_; ++o) {
        const float e = __expf(lg[o] - mx);
        lg[o] = e;
        ssum += e;
    }
    const float inv = 1.f / ssum;

    float acc[DH_] = {0.f, 0.f, 0.f, 0.f, 0.f, 0.f, 0.f, 0.f};
    {
        int o = 0;
#pragma unroll
        for (int di = 0; di < KS_; ++di) {
#pragma unroll
            for (int dj = 0; dj < KS_; ++dj, ++o) {
                const float* vp = vs_ + (((py + di) * EXT_) + (px + dj)) * DH_;
                const float w = lg[o];
#pragma unroll
                for (int c = 0; c < DH_; ++c) acc[c] += w * vp[c];
            }
        }
    }

    // out: (B, 64, H, W) channel-planar
    float* op = out + (size_t)b * COUT_ * PIX_PER_IMG + (size_t)(h * DH_) * PIX_PER_IMG +
                (size_t)(i0 + py) * HW_ + (j0 + px);
#pragma unroll
    for (int c = 0; c < DH_; ++c) op[(size_t)c * PIX_PER_IMG] = acc[c] * inv;
}

// ---------------------------------------------------------------------------
extern "C" void kernel_launch(void* const* d_in, const int* in_sizes, int n_in,
                              void* d_out, int out_size, void* d_ws, size_t ws_size,
                              hipStream_t stream)
{
    (void)in_sizes; (void)n_in; (void)out_size; (void)ws_size;

    const float* x    = (const float*)d_in[0];
    // d_in[1] = r : unused by the reference
    const float* w_q  = (const float*)d_in[2];
    const float* w_k  = (const float*)d_in[3];
    const float* w_v  = (const float*)d_in[4];
    const float* relh = (const float*)d_in[5];
    const float* relw = (const float*)d_in[6];

    float* q_ws = (float*)d_ws;                       // 8 MB each
    float* k_ws = q_ws + (size_t)TOTPIX * COUT_;
    float* v_ws = k_ws + (size_t)TOTPIX * COUT_;

    saconv_qkv_wmma<<<256, 256, 0, stream>>>(x, w_q, w_k, w_v, q_ws, k_ws, v_ws);
    saconv_attn<<<B_ * HEADS_ * 16, 256, 0, stream>>>(q_ws, k_ws, v_ws, relh, relw,
                                                      (float*)d_out);
}